// PointNet2Segmentation_52295521796454
// MI455X (gfx1250) — compile-verified
//
#include <hip/hip_runtime.h>

typedef __attribute__((ext_vector_type(2))) float v2f;
typedef __attribute__((ext_vector_type(8))) float v8f;

#define WMMA4(A, Bf, C) __builtin_amdgcn_wmma_f32_16x16x4_f32(false, A, false, Bf, (short)0, C, false, false)

// LDS pitch skew: make pitch mod 64 == 4 (or 12) so 16 rows map to 16 banks.
__host__ __device__ constexpr int pitchOf(int k) { return (k % 64 == 4) ? k : k + 4; }

// ---------------------------------------------------------------------------
// Weight prep: Wt[n*cinPad + k] = (k < cin) ? W[k*cout + n] : 0
// Transposed + K-padded so WMMA B-fragments are one aligned b64 load, unguarded.
// ---------------------------------------------------------------------------
__global__ void weight_prep_kernel(const float* __restrict__ W, int cin, int cout, int cinPad,
                                   float* __restrict__ Wt, int total)
{
    int i = blockIdx.x * blockDim.x + threadIdx.x;
    if (i >= total) return;
    int n = i / cinPad, k = i - n * cinPad;
    Wt[i] = (k < cin) ? W[(long)k * cout + n] : 0.0f;
}

// ---------------------------------------------------------------------------
// FPS: one block per batch. dist + pos cached in LDS. argmax tie -> lowest idx.
// ---------------------------------------------------------------------------
__global__ __launch_bounds__(256) void fps_kernel(const float* __restrict__ pos, long bstride,
                                                  int N, int m, int* __restrict__ sel)
{
    __shared__ float px[2048], py[2048], pz[2048], dist[2048];
    __shared__ float rv[256];
    __shared__ int   ri[256];
    const int b = blockIdx.x, tid = threadIdx.x;
    const float* p = pos + (long)b * bstride;
    int* s = sel + (long)b * m;

    for (int i = tid; i < N; i += 256) { px[i]=p[i*3]; py[i]=p[i*3+1]; pz[i]=p[i*3+2]; }
    __syncthreads();
    float x0 = px[0], y0 = py[0], z0 = pz[0];
    for (int i = tid; i < N; i += 256) {
        float dx=px[i]-x0, dy=py[i]-y0, dz=pz[i]-z0;
        dist[i] = dx*dx + dy*dy + dz*dz;
    }
    if (tid == 0) s[0] = 0;
    __syncthreads();

    for (int t = 1; t < m; ++t) {
        float bv = -1.0f; int bi = 0;
        for (int i = tid; i < N; i += 256) {
            float v = dist[i];
            if (v > bv) { bv = v; bi = i; }   // strict > keeps earliest index
        }
        rv[tid] = bv; ri[tid] = bi;
        __syncthreads();
        for (int sh = 128; sh > 0; sh >>= 1) {
            if (tid < sh) {
                float ov = rv[tid+sh]; int oi = ri[tid+sh];
                if (ov > rv[tid] || (ov == rv[tid] && oi < ri[tid])) { rv[tid]=ov; ri[tid]=oi; }
            }
            __syncthreads();
        }
        int I = ri[0];
        if (tid == 0) s[t] = I;
        float xi = px[I], yi = py[I], zi = pz[I];
        for (int i = tid; i < N; i += 256) {
            float dx=px[i]-xi, dy=py[i]-yi, dz=pz[i]-zi;
            float d2 = dx*dx + dy*dy + dz*dz;
            if (d2 < dist[i]) dist[i] = d2;
        }
        __syncthreads();
    }
}

__global__ void gather_pos_kernel(const float* __restrict__ src, long sbs,
                                  const int* __restrict__ sel, float* __restrict__ dst,
                                  int m, int total)
{
    int i = blockIdx.x * blockDim.x + threadIdx.x;
    if (i >= total) return;
    int b = i / m;
    int sidx = sel[i];
    const float* sp = src + (long)b * sbs + (long)sidx * 3;
    dst[(long)i*3+0] = sp[0]; dst[(long)i*3+1] = sp[1]; dst[(long)i*3+2] = sp[2];
}

// ---------------------------------------------------------------------------
// Ball query: one wave per query. Repeated lexicographic-(d2,idx) min selection
// == top_k(-d2) semantics (ascending d2, ties by lower index), capped at K=64.
// ---------------------------------------------------------------------------
__global__ __launch_bounds__(256) void ball_kernel(const float* __restrict__ posq, long qbs, int m,
                                                   const float* __restrict__ posc, long cbs, int N,
                                                   float r2, int* __restrict__ idx, int* __restrict__ cnt)
{
    const int lane = threadIdx.x & 31;
    const int q = blockIdx.x * (blockDim.x >> 5) + (threadIdx.x >> 5);
    const int b = blockIdx.z;
    if (q >= m) return;
    const float* pq = posq + (long)b*qbs + (long)q*3;
    const float* pc = posc + (long)b*cbs;
    int* iq = idx + ((long)b*m + q) * 64;
    float qx = pq[0], qy = pq[1], qz = pq[2];
    float lastd = -1.0f; int lasti = -1;
    int t = 0;
    for (; t < 64; ++t) {
        float bd = 3.0e37f; int bi = 0x7fffffff;
        for (int j = lane; j < N; j += 32) {
            float dx = pc[j*3]-qx, dy = pc[j*3+1]-qy, dz = pc[j*3+2]-qz;
            float d2 = dx*dx + dy*dy + dz*dz;
            bool ok = (d2 <= r2) && ((d2 > lastd) || (d2 == lastd && j > lasti));
            if (ok && ((d2 < bd) || (d2 == bd && j < bi))) { bd = d2; bi = j; }
        }
        for (int off = 16; off > 0; off >>= 1) {
            float od = __shfl_xor(bd, off, 32);
            int   oi = __shfl_xor(bi, off, 32);
            if (od < bd || (od == bd && oi < bi)) { bd = od; bi = oi; }
        }
        if (bd >= 3.0e37f) break;     // uniform across wave after reduction
        if (lane == 0) iq[t] = bi;
        lastd = bd; lasti = bi;
    }
    if (lane == 0) cnt[(long)b*m + q] = t;
}

// ---------------------------------------------------------------------------
// WMMA f32 16x16x4 MLP layer on an LDS tile of 64 rows (skewed pitch).
// 2-row-tile blocking: B fragment reused across 2 accumulators; software-
// pipelined K loop (next fragments loaded before current WMMAs issue).
// ---------------------------------------------------------------------------
__device__ __forceinline__ void mlp_layer(const float* __restrict__ in, int kExt, int inPitch,
                                          const float* __restrict__ Wt, const float* __restrict__ bias,
                                          int cout, float* __restrict__ outLds, int outPitch,
                                          int wave, int lane)
{
    const int colTiles = cout >> 4;
    const int njobs = 2 * colTiles;           // 64 rows = 2 row-pairs of 32
    const int nloc = lane & 15;
    const int khalf = (lane < 16) ? 0 : 2;
    for (int job = wave; job < njobs; job += 8) {
        int rp = job / colTiles, ct = job - rp * colTiles;
        int n  = (ct << 4) + nloc;
        const float* a0p = in + ((rp << 5) + nloc) * inPitch + khalf;
        const float* a1p = a0p + 16 * inPitch;
        const float* wp  = Wt + (long)n * kExt + khalf;
        v8f acc0 = {}, acc1 = {};
        v2f a0c = *(const v2f*)a0p;
        v2f a1c = *(const v2f*)a1p;
        v2f bc  = *(const v2f*)wp;
        for (int k0 = 4; k0 < kExt; k0 += 4) {
            v2f a0n = *(const v2f*)(a0p + k0);
            v2f a1n = *(const v2f*)(a1p + k0);
            v2f bn_ = *(const v2f*)(wp + k0);
            acc0 = WMMA4(a0c, bc, acc0);
            acc1 = WMMA4(a1c, bc, acc1);
            a0c = a0n; a1c = a1n; bc = bn_;
        }
        acc0 = WMMA4(a0c, bc, acc0);
        acc1 = WMMA4(a1c, bc, acc1);
        float bnv = bias[n];
        int rb0 = (rp << 5) + ((lane < 16) ? 0 : 8);
#pragma unroll
        for (int v = 0; v < 8; ++v) {
            float v0 = acc0[v] + bnv;
            float v1 = acc1[v] + bnv;
            outLds[(rb0 + v)*outPitch + n]      = v0 > 0.0f ? v0 : 0.0f;
            outLds[(rb0 + 16 + v)*outPitch + n] = v1 > 0.0f ? v1 : 0.0f;
        }
    }
}

// ---------------------------------------------------------------------------
// Fused SA module: gather cat(x_j, pos_j - pos_q) -> 3-layer MLP (WMMA, LDS
// resident) -> masked column max over the 64 neighbors. One block per query.
// ---------------------------------------------------------------------------
template<int CF, int CPAD, int C1, int C2, int C3>
__global__ __launch_bounds__(256) void sa_mlp_kernel(
    const float* __restrict__ xin,  long xbs,
    const float* __restrict__ posc, long cbs,
    const float* __restrict__ posq, long qbs, int m,
    const int* __restrict__ idx, const int* __restrict__ cnt,
    const float* __restrict__ Wt0, const float* __restrict__ b0,
    const float* __restrict__ Wt1, const float* __restrict__ b1,
    const float* __restrict__ Wt2, const float* __restrict__ b2,
    float* __restrict__ out, long obs)
{
    constexpr int PF = pitchOf(CPAD);
    constexpr int PA = pitchOf(C1);
    constexpr int PB = pitchOf(C2);
    __shared__ float s_feat[64*PF];
    __shared__ float s_a[64*PA];
    __shared__ float s_b[64*PB];
    const int b = blockIdx.z, q = blockIdx.x;
    const int tid = threadIdx.x, lane = tid & 31, wave = tid >> 5;
    const int c = cnt[(long)b*m + q];
    const int* iq = idx + ((long)b*m + q) * 64;
    const float* xi = xin  + (long)b*xbs;
    const float* pc = posc + (long)b*cbs;
    const float* pq = posq + (long)b*qbs + (long)q*3;
    const float qx = pq[0], qy = pq[1], qz = pq[2];

    for (int e = tid; e < 64*PF; e += 256) {
        int j = e / PF, col = e - j*PF;
        float v = 0.0f;
        if (j < c) {
            int id = iq[j];
            if      (col <  CF)   v = xi[(long)id*CF + col];
            else if (col == CF)   v = pc[id*3+0] - qx;
            else if (col == CF+1) v = pc[id*3+1] - qy;
            else if (col == CF+2) v = pc[id*3+2] - qz;
        }
        s_feat[e] = v;
    }
    __syncthreads();
    mlp_layer(s_feat, CPAD, PF, Wt0, b0, C1, s_a, PA, wave, lane);
    __syncthreads();
    mlp_layer(s_a, C1, PA, Wt1, b1, C2, s_b, PB, wave, lane);
    __syncthreads();

    // final layer fused with masked max-pool: one W stream drives 4 row-tile
    // accumulators (4 WMMA per 5 loads), pipelined.
    float* oq = out + (long)b*obs + (long)q*C3;
    const int colTiles = C3 >> 4;
    const int nloc = lane & 15;
    const int khalf = (lane < 16) ? 0 : 2;
    for (int ct = wave; ct < colTiles; ct += 8) {
        int n = (ct << 4) + nloc;
        const float* wp  = Wt2 + (long)n * C2 + khalf;
        const float* a0p = s_b + nloc * PB + khalf;
        const float* a1p = a0p + 16 * PB;
        const float* a2p = a0p + 32 * PB;
        const float* a3p = a0p + 48 * PB;
        v8f acc0 = {}, acc1 = {}, acc2 = {}, acc3 = {};
        v2f a0c = *(const v2f*)a0p;
        v2f a1c = *(const v2f*)a1p;
        v2f a2c = *(const v2f*)a2p;
        v2f a3c = *(const v2f*)a3p;
        v2f bc  = *(const v2f*)wp;
        for (int k0 = 4; k0 < C2; k0 += 4) {
            v2f a0n = *(const v2f*)(a0p + k0);
            v2f a1n = *(const v2f*)(a1p + k0);
            v2f a2n = *(const v2f*)(a2p + k0);
            v2f a3n = *(const v2f*)(a3p + k0);
            v2f bn_ = *(const v2f*)(wp + k0);
            acc0 = WMMA4(a0c, bc, acc0);
            acc1 = WMMA4(a1c, bc, acc1);
            acc2 = WMMA4(a2c, bc, acc2);
            acc3 = WMMA4(a3c, bc, acc3);
            a0c = a0n; a1c = a1n; a2c = a2n; a3c = a3n; bc = bn_;
        }
        acc0 = WMMA4(a0c, bc, acc0);
        acc1 = WMMA4(a1c, bc, acc1);
        acc2 = WMMA4(a2c, bc, acc2);
        acc3 = WMMA4(a3c, bc, acc3);

        float bnv = b2[n];
        float colmax = -3.0e38f;
        int rhalf = (lane < 16) ? 0 : 8;
#pragma unroll
        for (int v = 0; v < 8; ++v) {
            int r = rhalf + v;
            float v0 = acc0[v] + bnv; v0 = v0 > 0.0f ? v0 : 0.0f; v0 = (r      < c) ? v0 : -3.0e38f;
            float v1 = acc1[v] + bnv; v1 = v1 > 0.0f ? v1 : 0.0f; v1 = (r + 16 < c) ? v1 : -3.0e38f;
            float v2_ = acc2[v] + bnv; v2_ = v2_ > 0.0f ? v2_ : 0.0f; v2_ = (r + 32 < c) ? v2_ : -3.0e38f;
            float v3 = acc3[v] + bnv; v3 = v3 > 0.0f ? v3 : 0.0f; v3 = (r + 48 < c) ? v3 : -3.0e38f;
            colmax = fmaxf(colmax, fmaxf(fmaxf(v0, v1), fmaxf(v2_, v3)));
        }
        colmax = fmaxf(colmax, __shfl_xor(colmax, 16, 32));
        if (lane < 16) oq[n] = colmax;
    }
}

// ---------------------------------------------------------------------------
// Generic WMMA GEMM, 32x64 block per wave (2 row-tiles x 4 col-tiles):
// 8 WMMA per 6 b64 loads, software-pipelined. cout % 64 == 0, R % 32 == 0.
// Y = relu(X @ W + b); batch in grid.z.
// ---------------------------------------------------------------------------
__global__ __launch_bounds__(256) void gemm_relu_kernel(
    const float* __restrict__ X, long xbs, int cinPad,
    const float* __restrict__ Wt, const float* __restrict__ bias, int cout,
    float* __restrict__ Y, long ybs, int rowPairs)
{
    const int lane = threadIdx.x & 31, wave = threadIdx.x >> 5;
    const int rp = blockIdx.x * 8 + wave;
    if (rp >= rowPairs) return;                 // wave-uniform
    const int ct = blockIdx.y, b = blockIdx.z;
    const float* Xb = X + (long)b*xbs;
    float* Yb = Y + (long)b*ybs;
    const int nloc = lane & 15;
    const int n0 = (ct << 6) + nloc;            // 64-wide strip
    const int khalf = (lane < 16) ? 0 : 2;
    const float* a0p = Xb + (long)((rp << 5) + nloc) * cinPad + khalf;
    const float* a1p = a0p + 16 * cinPad;
    const float* w0p = Wt + (long)(n0     ) * cinPad + khalf;
    const float* w1p = Wt + (long)(n0 + 16) * cinPad + khalf;
    const float* w2p = Wt + (long)(n0 + 32) * cinPad + khalf;
    const float* w3p = Wt + (long)(n0 + 48) * cinPad + khalf;

    v8f acc[8] = {};
    v2f a0c = *(const v2f*)a0p, a1c = *(const v2f*)a1p;
    v2f b0c = *(const v2f*)w0p, b1c = *(const v2f*)w1p;
    v2f b2c = *(const v2f*)w2p, b3c = *(const v2f*)w3p;
    for (int k0 = 4; k0 < cinPad; k0 += 4) {
        v2f a0n = *(const v2f*)(a0p + k0);
        v2f a1n = *(const v2f*)(a1p + k0);
        v2f b0n = *(const v2f*)(w0p + k0);
        v2f b1n = *(const v2f*)(w1p + k0);
        v2f b2n = *(const v2f*)(w2p + k0);
        v2f b3n = *(const v2f*)(w3p + k0);
        acc[0] = WMMA4(a0c, b0c, acc[0]);
        acc[1] = WMMA4(a0c, b1c, acc[1]);
        acc[2] = WMMA4(a0c, b2c, acc[2]);
        acc[3] = WMMA4(a0c, b3c, acc[3]);
        acc[4] = WMMA4(a1c, b0c, acc[4]);
        acc[5] = WMMA4(a1c, b1c, acc[5]);
        acc[6] = WMMA4(a1c, b2c, acc[6]);
        acc[7] = WMMA4(a1c, b3c, acc[7]);
        a0c = a0n; a1c = a1n;
        b0c = b0n; b1c = b1n; b2c = b2n; b3c = b3n;
    }
    acc[0] = WMMA4(a0c, b0c, acc[0]);
    acc[1] = WMMA4(a0c, b1c, acc[1]);
    acc[2] = WMMA4(a0c, b2c, acc[2]);
    acc[3] = WMMA4(a0c, b3c, acc[3]);
    acc[4] = WMMA4(a1c, b0c, acc[4]);
    acc[5] = WMMA4(a1c, b1c, acc[5]);
    acc[6] = WMMA4(a1c, b2c, acc[6]);
    acc[7] = WMMA4(a1c, b3c, acc[7]);

    const int rb0 = (rp << 5) + ((lane < 16) ? 0 : 8);
#pragma unroll
    for (int h = 0; h < 2; ++h) {
#pragma unroll
        for (int s = 0; s < 4; ++s) {
            int n = n0 + (s << 4);
            float bn = bias[n];
            int rb = rb0 + h * 16;
#pragma unroll
            for (int v = 0; v < 8; ++v) {
                float val = acc[h*4 + s][v] + bn;
                val = val > 0.0f ? val : 0.0f;
                Yb[(long)(rb + v)*cout + n] = val;
            }
        }
    }
}

// ---------------------------------------------------------------------------
// kNN-3 (lexicographic ties like top_k) + normalized inverse-d2 weights.
// ---------------------------------------------------------------------------
__global__ void knn3_kernel(const float* __restrict__ pdst, long dbs, int Md,
                            const float* __restrict__ psrc, long sbs, int Ms,
                            int* __restrict__ idx3, float* __restrict__ w3, int total)
{
    int i = blockIdx.x * blockDim.x + threadIdx.x;
    if (i >= total) return;
    int b = i / Md, r = i - b*Md;
    const float* pd = pdst + (long)b*dbs + (long)r*3;
    const float* ps = psrc + (long)b*sbs;
    float dx0 = pd[0], dy0 = pd[1], dz0 = pd[2];
    float bd0 = 3e38f, bd1 = 3e38f, bd2 = 3e38f;
    int   bi0 = 0,     bi1 = 0,     bi2 = 0;
    for (int j = 0; j < Ms; ++j) {
        float ax = ps[j*3]-dx0, ay = ps[j*3+1]-dy0, az = ps[j*3+2]-dz0;
        float d2 = ax*ax + ay*ay + az*az;
        if      (d2 < bd0) { bd2=bd1; bi2=bi1; bd1=bd0; bi1=bi0; bd0=d2; bi0=j; }
        else if (d2 < bd1) { bd2=bd1; bi2=bi1; bd1=d2; bi1=j; }
        else if (d2 < bd2) { bd2=d2; bi2=j; }
    }
    float w0 = 1.0f/fmaxf(bd0, 1e-16f), w1 = 1.0f/fmaxf(bd1, 1e-16f), w2 = 1.0f/fmaxf(bd2, 1e-16f);
    float ws = w0 + w1 + w2;
    idx3[(long)i*3+0] = bi0; idx3[(long)i*3+1] = bi1; idx3[(long)i*3+2] = bi2;
    w3[(long)i*3+0] = w0/ws; w3[(long)i*3+1] = w1/ws; w3[(long)i*3+2] = w2/ws;
}

__global__ void interp3_kernel(const float* __restrict__ src, long sbs, int C, int Md,
                               const int* __restrict__ idx3, const float* __restrict__ w3,
                               float* __restrict__ dst, int total)
{
    int i = blockIdx.x * blockDim.x + threadIdx.x;
    if (i >= total) return;
    int c = i % C, rb = i / C;      // rb in [0, B*Md)
    int b = rb / Md;
    const float* s = src + (long)b*sbs;
    int   i0 = idx3[(long)rb*3],   i1 = idx3[(long)rb*3+1],   i2 = idx3[(long)rb*3+2];
    float w0 = w3[(long)rb*3],     w1 = w3[(long)rb*3+1],     w2 = w3[(long)rb*3+2];
    dst[i] = w0*s[(long)i0*C + c] + w1*s[(long)i1*C + c] + w2*s[(long)i2*C + c];
}

// ---------------------------------------------------------------------------
// Concatenation / reduction helpers
// ---------------------------------------------------------------------------
__global__ void build_x3_kernel(const float* __restrict__ x2, const float* __restrict__ pos2,
                                float* __restrict__ X, int total)   // [B*256, 260]
{
    int i = blockIdx.x * blockDim.x + threadIdx.x;
    if (i >= total) return;
    int col = i % 260, rb = i / 260;
    float v;
    if      (col < 256) v = x2[(long)rb*256 + col];
    else if (col < 259) v = pos2[(long)rb*3 + (col-256)];
    else                v = 0.0f;
    X[i] = v;
}

__global__ void colmax_kernel(const float* __restrict__ t, float* __restrict__ g, int total) // B*1024
{
    int i = blockIdx.x * blockDim.x + threadIdx.x;
    if (i >= total) return;
    int b = i / 1024, c = i % 1024;
    const float* tb = t + (long)b*256*1024;
    float mx = -3e38f;
    for (int r = 0; r < 256; ++r) mx = fmaxf(mx, tb[(long)r*1024 + c]);
    g[i] = mx;
}

__global__ void build_xfp3_kernel(const float* __restrict__ g, const float* __restrict__ x2,
                                  float* __restrict__ X, int total)  // [B*256, 1280]
{
    int i = blockIdx.x * blockDim.x + threadIdx.x;
    if (i >= total) return;
    int col = i % 1280, rb = i / 1280, b = rb >> 8;
    X[i] = (col < 1024) ? g[(long)b*1024 + col] : x2[(long)rb*256 + (col-1024)];
}

__global__ void build_xfp2_kernel(const float* __restrict__ itp, const float* __restrict__ x1,
                                  float* __restrict__ X, int total)  // [B*1024, 384]
{
    int i = blockIdx.x * blockDim.x + threadIdx.x;
    if (i >= total) return;
    int col = i % 384, rb = i / 384;
    X[i] = (col < 256) ? itp[(long)rb*256 + col] : x1[(long)rb*128 + (col-256)];
}

__global__ void build_xfp1_kernel(const float* __restrict__ itp, const float* __restrict__ x,
                                  float* __restrict__ X, int total)  // [B*2048, 132]
{
    int i = blockIdx.x * blockDim.x + threadIdx.x;
    if (i >= total) return;
    int col = i % 132, rb = i / 132;
    float v;
    if      (col < 128) v = itp[(long)rb*128 + col];
    else if (col < 131) v = x[(long)rb*3 + (col-128)];
    else                v = 0.0f;
    X[i] = v;
}

// ---------------------------------------------------------------------------
extern "C" void kernel_launch(void* const* d_in, const int* in_sizes, int n_in,
                              void* d_out, int out_size, void* d_ws, size_t ws_size,
                              hipStream_t stream)
{
    (void)in_sizes; (void)n_in; (void)out_size; (void)ws_size;
    const float* x   = (const float*)d_in[0];
    const float* pos = (const float*)d_in[1];
    const float *sa1W0=(const float*)d_in[2],  *sa1b0=(const float*)d_in[3],
                *sa1W1=(const float*)d_in[4],  *sa1b1=(const float*)d_in[5],
                *sa1W2=(const float*)d_in[6],  *sa1b2=(const float*)d_in[7];
    const float *sa2W0=(const float*)d_in[8],  *sa2b0=(const float*)d_in[9],
                *sa2W1=(const float*)d_in[10], *sa2b1=(const float*)d_in[11],
                *sa2W2=(const float*)d_in[12], *sa2b2=(const float*)d_in[13];
    const float *sa3W0=(const float*)d_in[14], *sa3b0=(const float*)d_in[15],
                *sa3W1=(const float*)d_in[16], *sa3b1=(const float*)d_in[17],
                *sa3W2=(const float*)d_in[18], *sa3b2=(const float*)d_in[19];
    const float *fp3W0=(const float*)d_in[20], *fp3b0=(const float*)d_in[21],
                *fp3W1=(const float*)d_in[22], *fp3b1=(const float*)d_in[23];
    const float *fp2W0=(const float*)d_in[24], *fp2b0=(const float*)d_in[25],
                *fp2W1=(const float*)d_in[26], *fp2b1=(const float*)d_in[27];
    const float *fp1W0=(const float*)d_in[28], *fp1b0=(const float*)d_in[29],
                *fp1W1=(const float*)d_in[30], *fp1b1=(const float*)d_in[31],
                *fp1W2=(const float*)d_in[32], *fp1b2=(const float*)d_in[33];

    const int B = 8, N = 2048, M1 = 1024, M2 = 256;

    size_t off = 0;
    auto allocf = [&](size_t n) -> float* {
        off = (off + 255) & ~(size_t)255;
        float* p = (float*)((char*)d_ws + off); off += n * sizeof(float); return p;
    };
    auto alloci = [&](size_t n) -> int* {
        off = (off + 255) & ~(size_t)255;
        int* p = (int*)((char*)d_ws + off); off += n * sizeof(int); return p;
    };

    // transposed + padded weights [cout, cinPad]
    auto prep = [&](const float* W, int cin, int cout, int cinPad) -> float* {
        float* Wt = allocf((size_t)cout * cinPad);
        int total = cout * cinPad;
        weight_prep_kernel<<<(total + 255) / 256, 256, 0, stream>>>(W, cin, cout, cinPad, Wt, total);
        return Wt;
    };
    float* sa1T0 = prep(sa1W0,    6,   64,    8);
    float* sa1T1 = prep(sa1W1,   64,   64,   64);
    float* sa1T2 = prep(sa1W2,   64,  128,   64);
    float* sa2T0 = prep(sa2W0,  131,  128,  132);
    float* sa2T1 = prep(sa2W1,  128,  128,  128);
    float* sa2T2 = prep(sa2W2,  128,  256,  128);
    float* sa3T0 = prep(sa3W0,  259,  256,  260);
    float* sa3T1 = prep(sa3W1,  256,  512,  256);
    float* sa3T2 = prep(sa3W2,  512, 1024,  512);
    float* fp3T0 = prep(fp3W0, 1280,  256, 1280);
    float* fp3T1 = prep(fp3W1,  256,  256,  256);
    float* fp2T0 = prep(fp2W0,  384,  256,  384);
    float* fp2T1 = prep(fp2W1,  256,  128,  256);
    float* fp1T0 = prep(fp1W0,  131,  128,  132);
    float* fp1T1 = prep(fp1W1,  128,  128,  128);
    float* fp1T2 = prep(fp1W2,  128,  128,  128);

    int*   sel1 = alloci((size_t)B*M1);
    int*   sel2 = alloci((size_t)B*M2);
    float* pos1 = allocf((size_t)B*M1*3);
    float* pos2 = allocf((size_t)B*M2*3);
    int*   idx1 = alloci((size_t)B*M1*64);  int* cnt1 = alloci((size_t)B*M1);
    int*   idx2 = alloci((size_t)B*M2*64);  int* cnt2 = alloci((size_t)B*M2);
    float* x1   = allocf((size_t)B*M1*128);
    float* x2   = allocf((size_t)B*M2*256);
    float* X3   = allocf((size_t)B*M2*260);
    float* t3a  = allocf((size_t)B*M2*256);
    float* t3b  = allocf((size_t)B*M2*512);
    float* t3c  = allocf((size_t)B*M2*1024);
    float* g    = allocf((size_t)B*1024);
    float* Xf3  = allocf((size_t)B*M2*1280);
    float* t4a  = allocf((size_t)B*M2*256);
    float* h3   = allocf((size_t)B*M2*256);
    int*   ik2  = alloci((size_t)B*M1*3);   float* wk2 = allocf((size_t)B*M1*3);
    float* in2  = allocf((size_t)B*M1*256);
    float* Xf2  = allocf((size_t)B*M1*384);
    float* t5a  = allocf((size_t)B*M1*256);
    float* h2   = allocf((size_t)B*M1*128);
    int*   ik1  = alloci((size_t)B*N*3);    float* wk1 = allocf((size_t)B*N*3);
    float* in1  = allocf((size_t)B*N*128);
    float* Xf1  = allocf((size_t)B*N*132);
    float* t6a  = allocf((size_t)B*N*128);
    float* t6b  = allocf((size_t)B*N*128);

    auto gemm = [&](const float* X_, long xbs, int cinPad,
                    const float* Wt, const float* bias, int cout,
                    float* Y, long ybs, int R) {
        int rowPairs = R >> 5;
        dim3 grid((rowPairs + 7) / 8, cout >> 6, B);
        gemm_relu_kernel<<<grid, 256, 0, stream>>>(X_, xbs, cinPad, Wt, bias, cout, Y, ybs, rowPairs);
    };

    // ---- SA1 ----
    fps_kernel<<<dim3(B), 256, 0, stream>>>(pos, (long)N*3, N, M1, sel1);
    gather_pos_kernel<<<(B*M1+255)/256, 256, 0, stream>>>(pos, (long)N*3, sel1, pos1, M1, B*M1);
    ball_kernel<<<dim3(M1/8,1,B), 256, 0, stream>>>(pos1, (long)M1*3, M1, pos, (long)N*3, N, 0.04f, idx1, cnt1);
    sa_mlp_kernel<3,8,64,64,128><<<dim3(M1,1,B), 256, 0, stream>>>(
        x, (long)N*3, pos, (long)N*3, pos1, (long)M1*3, M1, idx1, cnt1,
        sa1T0, sa1b0, sa1T1, sa1b1, sa1T2, sa1b2, x1, (long)M1*128);

    // ---- SA2 ----
    fps_kernel<<<dim3(B), 256, 0, stream>>>(pos1, (long)M1*3, M1, M2, sel2);
    gather_pos_kernel<<<(B*M2+255)/256, 256, 0, stream>>>(pos1, (long)M1*3, sel2, pos2, M2, B*M2);
    ball_kernel<<<dim3(M2/8,1,B), 256, 0, stream>>>(pos2, (long)M2*3, M2, pos1, (long)M1*3, M1, 0.16f, idx2, cnt2);
    sa_mlp_kernel<128,132,128,128,256><<<dim3(M2,1,B), 256, 0, stream>>>(
        x1, (long)M1*128, pos1, (long)M1*3, pos2, (long)M2*3, M2, idx2, cnt2,
        sa2T0, sa2b0, sa2T1, sa2b1, sa2T2, sa2b2, x2, (long)M2*256);

    // ---- SA3 (global) ----
    build_x3_kernel<<<(B*M2*260+255)/256, 256, 0, stream>>>(x2, pos2, X3, B*M2*260);
    gemm(X3,  (long)M2*260,  260, sa3T0, sa3b0, 256,  t3a, (long)M2*256,  M2);
    gemm(t3a, (long)M2*256,  256, sa3T1, sa3b1, 512,  t3b, (long)M2*512,  M2);
    gemm(t3b, (long)M2*512,  512, sa3T2, sa3b2, 1024, t3c, (long)M2*1024, M2);
    colmax_kernel<<<(B*1024+255)/256, 256, 0, stream>>>(t3c, g, B*1024);

    // ---- FP3 (k=1 interp from single global point == broadcast g) ----
    build_xfp3_kernel<<<(B*M2*1280+255)/256, 256, 0, stream>>>(g, x2, Xf3, B*M2*1280);
    gemm(Xf3, (long)M2*1280, 1280, fp3T0, fp3b0, 256, t4a, (long)M2*256, M2);
    gemm(t4a, (long)M2*256,  256,  fp3T1, fp3b1, 256, h3,  (long)M2*256, M2);

    // ---- FP2 ----
    knn3_kernel<<<(B*M1+255)/256, 256, 0, stream>>>(pos1, (long)M1*3, M1, pos2, (long)M2*3, M2, ik2, wk2, B*M1);
    interp3_kernel<<<(B*M1*256+255)/256, 256, 0, stream>>>(h3, (long)M2*256, 256, M1, ik2, wk2, in2, B*M1*256);
    build_xfp2_kernel<<<(B*M1*384+255)/256, 256, 0, stream>>>(in2, x1, Xf2, B*M1*384);
    gemm(Xf2, (long)M1*384, 384, fp2T0, fp2b0, 256, t5a, (long)M1*256, M1);
    gemm(t5a, (long)M1*256, 256, fp2T1, fp2b1, 128, h2,  (long)M1*128, M1);

    // ---- FP1 ----
    knn3_kernel<<<(B*N+255)/256, 256, 0, stream>>>(pos, (long)N*3, N, pos1, (long)M1*3, M1, ik1, wk1, B*N);
    interp3_kernel<<<(B*N*128+255)/256, 256, 0, stream>>>(h2, (long)M1*128, 128, N, ik1, wk1, in1, B*N*128);
    build_xfp1_kernel<<<(B*N*132+255)/256, 256, 0, stream>>>(in1, x, Xf1, B*N*132);
    gemm(Xf1, (long)N*132, 132, fp1T0, fp1b0, 128, t6a, (long)N*128, N);
    gemm(t6a, (long)N*128, 128, fp1T1, fp1b1, 128, t6b, (long)N*128, N);
    gemm(t6b, (long)N*128, 128, fp1T2, fp1b2, 128, (float*)d_out, (long)N*128, N);
}